// SoftBellmanFord_5471788335893
// MI455X (gfx1250) — compile-verified
//
#include <hip/hip_runtime.h>
#include <stdint.h>

// Soft Bellman-Ford: 50 iterations of R = clip(R + 0.25*(4-neighbor sum)*P, 0, 1)
// on (32,1024,1024) f32, then gather at goal coords.
//
// Strategy (MI455X): memory-bound stencil (0.5 flop/byte) -> temporal blocking.
//   - Ghost-padded (1056x1056) copies of P and R in d_ws (ghost ring = 0
//     emulates 'SAME' zero padding, and makes every tile load unclipped).
//   - Each workgroup owns a 160x160 tile (128x128 valid + 16 halo), loads
//     P-tile and R-tile into LDS with TWO Tensor-Data-Mover descriptors
//     (tensor_load_to_lds, s_wait_tensorcnt), runs 16 iterations ping-pong
//     in LDS (300KB of the 320KB WGP LDS), writes the central 128x128 back.
//   - 50 = 16+16+16+2 -> 4 passes, ~2.2 GB total HBM traffic vs 19.2 GB naive.

#define HH   1024
#define WW   1024
#define NB   32
#define PAD  16
#define PW   1056                 // padded width/height = 1024 + 2*16
#define PW4  (PW / 4)             // 264 float4 chunks per padded row
#define TILE 160                  // tile with halo
#define CENT 128                  // central (valid) region
#define LDS_TILE (TILE * TILE)    // 25600 floats = 102400 bytes

typedef __attribute__((ext_vector_type(4))) unsigned int v4u;
typedef __attribute__((ext_vector_type(8))) int          v8i;
typedef __attribute__((ext_vector_type(4))) int          v4i;
typedef __attribute__((ext_vector_type(4))) float        f4;

// ---------------------------------------------------------------- init ----
// Build ghost-padded P, zero both R ping-pong buffers (ghost ring must be 0
// every call: harness poisons d_ws and never re-initializes between replays).
// Division-free: block = one padded row of one batch, thread = one f4 chunk.
// PAD==16 is f4-aligned, so each chunk is entirely ghost or entirely interior.
__global__ void sbf_init(const float* __restrict__ P,
                         float* __restrict__ Ppad,
                         float* __restrict__ Ra,
                         float* __restrict__ Rb)
{
    int t = threadIdx.x;
    if (t >= PW4) return;
    int y = blockIdx.x;            // padded row   [0,1056)
    int b = blockIdx.y;            // batch        [0,32)
    int x4 = t * 4;                // padded col (f4 granularity)

    f4 pv = {0.0f, 0.0f, 0.0f, 0.0f};
    if (y >= PAD && y < PAD + HH && x4 >= PAD && x4 < PAD + WW)
        pv = *(const f4*)(P + (long long)b * HH * WW
                            + (long long)(y - PAD) * WW + (x4 - PAD));

    long long i = (long long)b * PW * PW + (long long)y * PW + x4;
    f4 z = {0.0f, 0.0f, 0.0f, 0.0f};
    *(f4*)(Ppad + i) = pv;
    *(f4*)(Ra + i)   = z;
    *(f4*)(Rb + i)   = z;
}

__global__ void sbf_seed(const int* __restrict__ start, float* __restrict__ Ra)
{
    int b = threadIdx.x;
    if (b >= NB) return;
    int sr = min(max(start[2 * b + 0], 0), HH - 1);
    int sc = min(max(start[2 * b + 1], 0), WW - 1);
    Ra[(long long)b * PW * PW + (long long)(sr + PAD) * PW + (sc + PAD)] = 1.0f;
}

// ---------------------------------------------------------------- pass ----
__global__ __launch_bounds__(1024, 1)
void sbf_pass(const float* __restrict__ Ppad,
              const float* __restrict__ Rsrc,
              float* __restrict__ Rdst,
              int iters)
{
    extern __shared__ __align__(16) float smem[];   // 3 * 102400 B = 300 KB
    float* sP = smem;                 // LDS byte offset 0
    float* sA = smem + LDS_TILE;      // LDS byte offset 102400
    float* sB = smem + 2 * LDS_TILE;  // LDS byte offset 204800

    const int tx = blockIdx.x, ty = blockIdx.y, b = blockIdx.z;
    // Halo origin of this tile inside the padded array (always in-bounds).
    const long long origin = (long long)b * PW * PW
                           + (long long)(ty * CENT) * PW + (long long)(tx * CENT);

    // ---- TDM: one wave issues two 2D tensor loads (160x160 f32, stride 1056,
    // packed contiguously into LDS since LDS row stride == tile width). ----
    if (threadIdx.x == 0) {
        // D# group1 (shared by both loads): data_size=4B, tensor_dim0=1056,
        // tensor_dim1=1056, tile_dim0=160, tile_dim1=160, dim0_stride=1056.
        v8i g1;
        g1[0] = (int)(2u << 16);                       // data_size = 2 (4 bytes)
        g1[1] = (int)(((unsigned)PW & 0xFFFFu) << 16); // tensor_dim0[15:0]
        g1[2] = (int)(((unsigned)PW & 0xFFFFu) << 16); // tensor_dim1[15:0]
        g1[3] = (int)(((unsigned)TILE & 0xFFFFu) << 16); // tile_dim0
        g1[4] = (int)((unsigned)TILE & 0xFFFFu);       // tile_dim1 (tile_dim2=0)
        g1[5] = (int)(unsigned)PW;                     // tensor_dim0_stride[31:0]
        g1[6] = 0;
        g1[7] = 0;
        v4i gz4 = {0, 0, 0, 0};                        // 2D tensor: groups 2/3 zero
        v8i gz8 = {0, 0, 0, 0, 0, 0, 0, 0};            // trailing group (6-arg form)

        uint64_t gaP = (uint64_t)(Ppad + origin);
        uint64_t gaR = (uint64_t)(Rsrc + origin);

        v4u g0;
        g0[0] = 1u;                                    // count=1 (valid), no gather
        g0[1] = 0u;                                    // lds_addr = sP (offset 0)
        g0[2] = (unsigned)gaP;                         // global_addr[31:0]
        g0[3] = ((unsigned)(gaP >> 32) & 0x01FFFFFFu) | 0x80000000u; // addr[56:32] | type=2
        __builtin_amdgcn_tensor_load_to_lds(g0, g1, gz4, gz4, gz8, 0);

        g0[1] = (unsigned)(LDS_TILE * 4);              // lds_addr = sA
        g0[2] = (unsigned)gaR;
        g0[3] = ((unsigned)(gaR >> 32) & 0x01FFFFFFu) | 0x80000000u;
        __builtin_amdgcn_tensor_load_to_lds(g0, g1, gz4, gz4, gz8, 0);

        __builtin_amdgcn_s_wait_tensorcnt(0);
    }
    __syncthreads();

    // ---- temporal iterations, ping-pong in LDS -----------------------------
    // Update rows y in [1,158], all 40 float4 column chunks (ring columns get
    // garbage but are never consumed: a cell at distance d from the tile edge
    // is exact for d iterations; the central region (d>=16) survives 16 iters.
    float* s0 = sA;
    float* s1 = sB;
    for (int it = 0; it < iters; ++it) {
        for (int w = threadIdx.x; w < 40 * 158; w += 1024) {
            int y   = (w / 40) + 1;
            int c   = (w % 40) * 4;
            int idx = y * TILE + c;
            f4 cen = *(const f4*)(s0 + idx);
            f4 up  = *(const f4*)(s0 + idx - TILE);
            f4 dn  = *(const f4*)(s0 + idx + TILE);
            f4 lft = *(const f4*)(s0 + idx - 4);   // in-bounds: wraps to prev row
            f4 rgt = *(const f4*)(s0 + idx + 4);   // in-bounds: wraps to next row
            f4 p   = *(const f4*)(sP + idx);
            f4 lv, rv;
            lv[0] = lft[3]; lv[1] = cen[0]; lv[2] = cen[1]; lv[3] = cen[2];
            rv[0] = cen[1]; rv[1] = cen[2]; rv[2] = cen[3]; rv[3] = rgt[0];
            f4 n = (up + dn + lv + rv) * 0.25f;
            f4 r = cen + n * p;
#pragma unroll
            for (int k = 0; k < 4; ++k) r[k] = fminf(fmaxf(r[k], 0.0f), 1.0f);
            *(f4*)(s1 + idx) = r;
        }
        __syncthreads();
        float* t = s0; s0 = s1; s1 = t;
    }

    // ---- write back central 128x128 (b128 stores, 16B aligned both sides) --
    for (int w = threadIdx.x; w < (CENT * CENT) / 4; w += 1024) {
        int i = w / (CENT / 4);
        int j = (w % (CENT / 4)) * 4;
        f4 v = *(const f4*)(s0 + (PAD + i) * TILE + PAD + j);
        *(f4*)(Rdst + origin + (long long)(PAD + i) * PW + (PAD + j)) = v;
    }
}

// -------------------------------------------------------------- gather ----
__global__ void sbf_gather(const int* __restrict__ goal,
                           const float* __restrict__ Rfin,
                           float* __restrict__ out)
{
    int b = threadIdx.x;
    if (b >= NB) return;
    int gr = min(max(goal[2 * b + 0], 0), HH - 1);
    int gc = min(max(goal[2 * b + 1], 0), WW - 1);
    out[b] = Rfin[(long long)b * PW * PW + (long long)(gr + PAD) * PW + (gc + PAD)];
}

// -------------------------------------------------------------- launch ----
extern "C" void kernel_launch(void* const* d_in, const int* in_sizes, int n_in,
                              void* d_out, int out_size, void* d_ws, size_t ws_size,
                              hipStream_t stream)
{
    (void)in_sizes; (void)n_in; (void)out_size; (void)ws_size;
    const float* P     = (const float*)d_in[0];
    const int*   start = (const int*)d_in[1];
    const int*   goal  = (const int*)d_in[2];
    float*       out   = (float*)d_out;

    // Workspace layout: Ppad | Ra | Rb, each 32 * 1056*1056 f32 (~143 MB each).
    const size_t plane = (size_t)PW * PW;
    float* Ppad = (float*)d_ws;
    float* Ra   = Ppad + (size_t)NB * plane;
    float* Rb   = Ra   + (size_t)NB * plane;

    // Init: one block per (padded row, batch); 288 threads cover 264 f4 chunks.
    sbf_init<<<dim3(PW, NB), 288, 0, stream>>>(P, Ppad, Ra, Rb);
    sbf_seed<<<1, 32, 0, stream>>>(start, Ra);

    dim3 grid(WW / CENT, HH / CENT, NB);              // 8 x 8 x 32 tiles
    size_t shmem = 3 * LDS_TILE * sizeof(float);      // 307200 B in 320KB WGP LDS
    sbf_pass<<<grid, 1024, shmem, stream>>>(Ppad, Ra, Rb, 16);
    sbf_pass<<<grid, 1024, shmem, stream>>>(Ppad, Rb, Ra, 16);
    sbf_pass<<<grid, 1024, shmem, stream>>>(Ppad, Ra, Rb, 16);
    sbf_pass<<<grid, 1024, shmem, stream>>>(Ppad, Rb, Ra, 2);   // 16*3 + 2 = 50

    sbf_gather<<<1, 32, 0, stream>>>(goal, Ra, out);
}